// VectorQuantizeImage_34359739043
// MI455X (gfx1250) — compile-verified
//
#include <hip/hip_runtime.h>

typedef float v2f __attribute__((ext_vector_type(2)));
typedef float v8f __attribute__((ext_vector_type(8)));
typedef unsigned int u32x4 __attribute__((ext_vector_type(4)));
typedef int i32x4 __attribute__((ext_vector_type(4)));
typedef int i32x8 __attribute__((ext_vector_type(8)));

#define K_CODES 1024
#define D_DIM   64
#define CHUNK   128      // codes staged per TDM transfer
#define NCHUNK  (K_CODES / CHUNK)
#define LSTRIDE 68       // 64 data + 4 pad floats (TDM pad_interval=64dw, pad_amount=4dw)
#define ROWS_PER_BLOCK 128
#define THREADS 256

// ---- LDS byte-offset of a __shared__ object (ptrtoint of addrspace(3) ptr) ----
static __device__ __forceinline__ unsigned lds_off(const void* p) {
    return (unsigned)(uintptr_t)(__attribute__((address_space(3))) const void*)p;
}

// ---- issue one TDM transfer: CHUNK codebook rows -> padded LDS tile ----
// D# group0: count=1, lds_addr, global_addr(57b), type=2
// D# group1: data_size=4B, pad_enable, pad_interval=64 DWORDs, pad_amount=4 DWORDs,
//            tensor_dim0=64, tensor_dim1=1024, tile_dim0=64, tile_dim1=128, stride0=64
static __device__ __forceinline__ void tdm_stage_chunk(const float* gsrc, unsigned lds_byte) {
    unsigned long long va = (unsigned long long)(uintptr_t)gsrc;
    u32x4 g0;
    g0[0] = 1u;                                                // count=1, no gather
    g0[1] = lds_byte;                                          // lds_addr
    g0[2] = (unsigned)(va & 0xffffffffu);                      // global_addr[31:0]
    g0[3] = (unsigned)((va >> 32) & 0x01ffffffu) | (2u << 30); // addr[56:32] | type=2
    i32x8 g1;
    g1[0] = (2 << 16) | (1 << 20) | (5 << 22) | (3 << 25);     // dsz=4B|pad_en|int=64dw|amt=4dw
    g1[1] = (D_DIM & 0xffff) << 16;                            // tensor_dim0 = 64
    g1[2] = (K_CODES & 0xffff) << 16;                          // tensor_dim1 = 1024 (lo16)
    g1[3] = (D_DIM & 0xffff) << 16;                            // tile_dim0 = 64
    g1[4] = CHUNK;                                             // tile_dim1 = 128, tile_dim2 = 0
    g1[5] = D_DIM;                                             // tensor_dim0_stride = 64
    g1[6] = 0;
    g1[7] = 0;
    i32x4 z4 = {0, 0, 0, 0};
    i32x8 z8 = {0, 0, 0, 0, 0, 0, 0, 0};
    __builtin_amdgcn_tensor_load_to_lds(g0, g1, z4, z4, z8, 0);
}

// ---------------- kernel 1: ||w||^2 per code ----------------
__global__ __launch_bounds__(256)
void wnorm_kernel(const float* __restrict__ W, float* __restrict__ wnorm) {
    int k = blockIdx.x * blockDim.x + threadIdx.x;
    if (k >= K_CODES) return;
    const float4* w4 = (const float4*)(W + (size_t)k * D_DIM);
    float s = 0.f;
#pragma unroll
    for (int i = 0; i < D_DIM / 4; ++i) {
        float4 v = w4[i];
        s += v.x * v.x + v.y * v.y + v.z * v.z + v.w * v.w;
    }
    wnorm[k] = s;
}

// ---------------- kernel 2: fused distance + argmin (fp32 WMMA + TDM pipeline) ----------------
__global__ __launch_bounds__(THREADS)
void vq_argmin_kernel(const float* __restrict__ X, const float* __restrict__ W,
                      const float* __restrict__ wnorm, int* __restrict__ idx_out,
                      int nrows) {
    __shared__ float lw[2][CHUNK * LSTRIDE];   // double-buffered codebook tiles
    __shared__ float lwn[K_CODES];             // ||w||^2, staged once

    const int tid  = threadIdx.x;
    const int wave = tid >> 5;
    const int lane = tid & 31;
    const int half = lane >> 4;
    const int lr   = lane & 15;
    const int off  = 2 * half;
    const int rowBase = blockIdx.x * ROWS_PER_BLOCK + wave * 16;

    // stage wnorm into LDS (visible after the first chunk barrier)
    for (int i = tid; i < K_CODES; i += THREADS) lwn[i] = wnorm[i];

    // A tile: 16 rows x 64 dims, scaled by -2, resident in 32 VGPRs
    int arow = rowBase + lr;
    if (arow >= nrows) arow = nrows - 1;
    const float* xrow = X + (size_t)arow * D_DIM;
    v2f a[16];
#pragma unroll
    for (int j = 0; j < 16; ++j) {
        a[j].x = -2.0f * xrow[4 * j + off];
        a[j].y = -2.0f * xrow[4 * j + off + 1];
    }

    float bestd[8];
    int   bestc[8];
#pragma unroll
    for (int i = 0; i < 8; ++i) { bestd[i] = 3.4e38f; bestc[i] = 0; }

    // TDM prologue: wave 0 kicks chunk 0 into buffer 0
    if (wave == 0) tdm_stage_chunk(W, lds_off(&lw[0][0]));

    int cur = 0;
#pragma unroll 1
    for (int chunk = 0; chunk < NCHUNK; ++chunk) {
        if (wave == 0) __builtin_amdgcn_s_wait_tensorcnt(0);  // buf[cur] DMA complete
        __syncthreads();                                       // all waves: buf[cur] ready,
                                                               // buf[cur^1] fully consumed
        if (wave == 0 && chunk + 1 < NCHUNK)
            tdm_stage_chunk(W + (size_t)(chunk + 1) * CHUNK * D_DIM,
                            lds_off(&lw[cur ^ 1][0]));         // overlap DMA with compute

        const float* buf = &lw[cur][0];
#pragma unroll
        for (int sub = 0; sub < CHUNK / 16; ++sub) {
            const int codeL = sub * 16 + lr;
            const int code  = chunk * CHUNK + codeL;
            const float* bp = buf + codeL * LSTRIDE + off;

            // batch the B-tile LDS reads, then run the WMMA chain back-to-back
            v2f b[16];
#pragma unroll
            for (int j = 0; j < 16; ++j) { b[j].x = bp[4 * j]; b[j].y = bp[4 * j + 1]; }
            const float wn = lwn[code];

            v8f c = {};
#pragma unroll
            for (int j = 0; j < 16; ++j)
                c = __builtin_amdgcn_wmma_f32_16x16x4_f32(
                        false, a[j], false, b[j], (short)0, c, false, false);

#pragma unroll
            for (int i = 0; i < 8; ++i) {
                float d  = wn + c[i];
                bool upd = d < bestd[i];
                bestd[i] = upd ? d : bestd[i];
                bestc[i] = upd ? code : bestc[i];
            }
        }
        cur ^= 1;
    }

    // reduce each C-row across its 16 candidate lanes (within each half-wave)
#pragma unroll
    for (int i = 0; i < 8; ++i) {
        float d = bestd[i];
        int   cc = bestc[i];
#pragma unroll
        for (int m = 8; m >= 1; m >>= 1) {
            float od = __shfl_xor(d, m, 32);
            int   oc = __shfl_xor(cc, m, 32);
            bool  t  = (od < d) || (od == d && oc < cc);
            d  = t ? od : d;
            cc = t ? oc : cc;
        }
        if (lr == 0) {
            int row = rowBase + half * 8 + i;
            if (row < nrows) idx_out[row] = cc;
        }
    }
}

// ---------------- kernel 3: gather quantized rows + histogram ----------------
__global__ __launch_bounds__(256)
void gather_hist_kernel(const float* __restrict__ W, const int* __restrict__ idx,
                        float* __restrict__ quant, float* __restrict__ idxf,
                        int* __restrict__ counts, int nrows) {
    __shared__ int sidx[256];
    __shared__ int shist[K_CODES];
    const int tid = threadIdx.x;

    for (int i = tid; i < K_CODES; i += 256) shist[i] = 0;
    __syncthreads();

    const int row = blockIdx.x * 256 + tid;
    int my = 0;
    if (row < nrows) {
        my = idx[row];
        idxf[row] = (float)my;
        atomicAdd(&shist[my], 1);
    }
    sidx[tid] = my;
    __syncthreads();

    const int base = blockIdx.x * 256;
    for (int s = tid; s < 256 * (D_DIM / 4); s += 256) {
        int r    = s >> 4;
        int comp = s & 15;
        int grow = base + r;
        if (grow < nrows) {
            const float4* src = (const float4*)(W + (size_t)sidx[r] * D_DIM);
            ((float4*)(quant + (size_t)grow * D_DIM))[comp] = src[comp];
        }
    }
    __syncthreads();

    for (int i = tid; i < K_CODES; i += 256) {
        int v = shist[i];
        if (v) atomicAdd(&counts[i], v);
    }
}

// ---------------- kernel 4: entropy / perplexity / diversity ----------------
__global__ __launch_bounds__(1024)
void stats_kernel(const int* __restrict__ counts, float* __restrict__ outPerp,
                  float* __restrict__ outDiv, float invN) {
    __shared__ float red[32];
    const int tid = threadIdx.x;
    float p = (float)counts[tid] * invN;
    float e = -p * logf(p + 1e-10f);
#pragma unroll
    for (int m = 16; m >= 1; m >>= 1) e += __shfl_xor(e, m, 32);
    if ((tid & 31) == 0) red[tid >> 5] = e;
    __syncthreads();
    if (tid < 32) {
        float v = red[tid];
#pragma unroll
        for (int m = 16; m >= 1; m >>= 1) v += __shfl_xor(v, m, 32);
        if (tid == 0) {
            float H = v;
            *outPerp = expf(H);
            *outDiv  = 1.0f - H / logf((float)K_CODES);
        }
    }
}

// ---------------- launcher ----------------
extern "C" void kernel_launch(void* const* d_in, const int* in_sizes, int n_in,
                              void* d_out, int out_size, void* d_ws, size_t ws_size,
                              hipStream_t stream) {
    const float* X = (const float*)d_in[0];           // [N, 64] flattened
    const float* W = (const float*)d_in[1];           // [1024, 64]
    const int n = in_sizes[0] / D_DIM;                // 131072

    float* out   = (float*)d_out;
    float* quant = out;                               // n * 64
    float* idxf  = out + (size_t)n * D_DIM;           // n
    float* perp  = idxf + n;                          // 1
    float* divl  = perp + 1;                          // 1

    float* wnorm  = (float*)d_ws;                     // 1024 f32
    int*   counts = (int*)((char*)d_ws + 4096);       // 1024 i32
    int*   idx    = (int*)((char*)d_ws + 8192);       // n i32

    (void)hipMemsetAsync(counts, 0, K_CODES * sizeof(int), stream);
    wnorm_kernel<<<K_CODES / 256, 256, 0, stream>>>(W, wnorm);

    const int nblocks = (n + ROWS_PER_BLOCK - 1) / ROWS_PER_BLOCK;
    vq_argmin_kernel<<<nblocks, THREADS, 0, stream>>>(X, W, wnorm, idx, n);

    gather_hist_kernel<<<(n + 255) / 256, 256, 0, stream>>>(W, idx, quant, idxf, counts, n);
    stats_kernel<<<1, K_CODES, 0, stream>>>(counts, perp, divl, 1.0f / (float)n);
}